// DynamicGraphLayer_48842368090160
// MI455X (gfx1250) — compile-verified
//
#include <hip/hip_runtime.h>
#include <hip/hip_bf16.h>
#include <math.h>
#include <stdint.h>

typedef __attribute__((ext_vector_type(2))) float v2f;
typedef __attribute__((ext_vector_type(8))) float v8f;

#define NODE_DIM 128
#define EDGE_DIM 64
#define HID 128

static __device__ __forceinline__ v8f wmma4(v2f a, v2f b, v8f c) {
  // V_WMMA_F32_16X16X4_F32 : D = A(16x4) * B(4x16) + C(16x16), f32
  return __builtin_amdgcn_wmma_f32_16x16x4_f32(
      /*neg_a=*/false, a, /*neg_b=*/false, b,
      /*c_mod=*/(short)0, c, /*reuse_a=*/false, /*reuse_b=*/false);
}

// Stage `nfloats` (multiple of 4*nthreads) from global into LDS using the
// gfx1250 async global->LDS DMA path (ASYNCcnt), 16B per lane per issue.
static __device__ __forceinline__ void stage_lds_async(const float* __restrict__ g,
                                                       float* s, int nfloats,
                                                       int tid, int nthreads) {
  const int chunks = nfloats >> 2;            // 16-byte chunks
  for (int i = tid; i < chunks; i += nthreads) {
    uint32_t lds_addr = (uint32_t)(uintptr_t)(s + (i << 2));  // low 32b = LDS byte offset
    const float* ga = g + (i << 2);
    asm volatile("global_load_async_to_lds_b128 %0, %1, off"
                 :: "v"(lds_addr), "v"(ga) : "memory");
  }
  asm volatile("s_wait_asynccnt 0x0" ::: "memory");  // this wave's DMAs landed in LDS
}

// ---------------- elementwise helpers ----------------

__global__ void k_set1(float* __restrict__ p, int n) {
  int i = blockIdx.x * blockDim.x + threadIdx.x;
  if (i < n) p[i] = 1.0f;                       // self-loop contributes 1 to deg
}

__global__ void k_count_deg(const int* __restrict__ col, float* __restrict__ deg, int e) {
  int i = blockIdx.x * blockDim.x + threadIdx.x;
  if (i < e) atomicAdd(&deg[col[i]], 1.0f);
}

__global__ void k_rsqrt_inplace(float* __restrict__ p, int n) {
  int i = blockIdx.x * blockDim.x + threadIdx.x;
  if (i < n) p[i] = rsqrtf(p[i]);               // deg >= 1 always (self loop)
}

// acc[i,c] = bias[c] + hpre[i,c] * dinv[i]^2   (self-loop message + conv bias)
// float4 per thread: total4 = n*128/4
__global__ void k_init_acc4(const float* __restrict__ hpre, const float* __restrict__ dinv,
                            const float* __restrict__ bias, float* __restrict__ acc,
                            int total4) {
  int idx = blockIdx.x * blockDim.x + threadIdx.x;
  if (idx >= total4) return;
  int i = idx >> 5;                      // node: (idx*4)/128
  int cb = (idx & 31) << 2;              // channel base
  float d = dinv[i];
  float d2 = d * d;
  float4 h = *(const float4*)(hpre + ((size_t)idx << 2));
  float4 b = *(const float4*)(bias + cb);
  float4 r;
  r.x = b.x + h.x * d2; r.y = b.y + h.y * d2;
  r.z = b.z + h.z * d2; r.w = b.w + h.w * d2;
  *(float4*)(acc + ((size_t)idx << 2)) = r;
}

static __device__ __forceinline__ float gelu_exact(float x) {
  return 0.5f * x * (1.0f + erff(x * 0.70710678118654752f));
}

__global__ void k_gelu_inplace4(float* __restrict__ p, int total4) {
  int i = blockIdx.x * blockDim.x + threadIdx.x;
  if (i >= total4) return;
  float4 v = *(const float4*)(p + ((size_t)i << 2));
  v.x = gelu_exact(v.x); v.y = gelu_exact(v.y);
  v.z = gelu_exact(v.z); v.w = gelu_exact(v.w);
  *(float4*)(p + ((size_t)i << 2)) = v;
}

// ---------------- f32 WMMA GEMM: C[M x 128] = A[M x K] @ B[K x 128] ----------------
// Block = 128 threads (4 waves). One block computes 16 rows; wave w owns cols [32w,32w+32).
template <int K>
__global__ __launch_bounds__(128) void k_gemm_wmma(const float* __restrict__ A,
                                                   const float* __restrict__ B,
                                                   float* __restrict__ C, int M) {
  __shared__ float sB[K * 128];
  stage_lds_async(B, sB, K * 128, threadIdx.x, 128);
  __syncthreads();

  const int wave = threadIdx.x >> 5;
  const int lane = threadIdx.x & 31;
  const int m0   = blockIdx.x * 16;
  const int mrow = lane & 15;
  const int hi   = lane >> 4;          // 0 for lanes 0-15, 1 for lanes 16-31
  const int kh   = hi * 2;             // K-offset of this lane-half
  const int n0   = wave * 32;
  const int ncol = lane & 15;

  int arow_i = m0 + mrow; if (arow_i >= M) arow_i = M - 1;   // clamp (tail safety)
  const float* __restrict__ arow = A + (size_t)arow_i * K;

  v8f c0 = {}; v8f c1 = {};
#pragma unroll 4
  for (int k = 0; k < K; k += 4) {
    v2f a = *(const v2f*)(arow + k + kh);                    // A[m][k+kh], A[m][k+kh+1]
    v2f b0, b1;
    b0.x = sB[(k + kh    ) * 128 + n0 + ncol];
    b0.y = sB[(k + kh + 1) * 128 + n0 + ncol];
    b1.x = sB[(k + kh    ) * 128 + n0 + 16 + ncol];
    b1.y = sB[(k + kh + 1) * 128 + n0 + 16 + ncol];
    c0 = wmma4(a, b0, c0);
    c1 = wmma4(a, b1, c1);
  }

  // C/D layout: VGPR v holds row (v + 8*hi), col = n0(+16) + ncol
  const int rbase = m0 + (hi << 3);
#pragma unroll
  for (int v = 0; v < 8; ++v) {
    int r = rbase + v;
    if (r < M) {
      C[(size_t)r * 128 + n0 + ncol]      = c0[v];
      C[(size_t)r * 128 + n0 + 16 + ncol] = c1[v];
    }
  }
}

// ---------------- fused edge kernel ----------------
// For a tile of 16 edges: ep = edge_attr[e0:e0+16] @ We (64x128) via WMMA, then
// atomically acc[col[e]] += (ep + hpre[row[e]]) * dinv[row]*dinv[col].
__global__ __launch_bounds__(128) void k_edge_msg(const float* __restrict__ edge_attr,
                                                  const float* __restrict__ We,
                                                  const int* __restrict__ row,
                                                  const int* __restrict__ col,
                                                  const float* __restrict__ hpre,
                                                  const float* __restrict__ dinv,
                                                  float* __restrict__ acc, int E) {
  __shared__ float sW[EDGE_DIM * 128];   // 32 KB
  __shared__ int   sRow[16], sCol[16];
  __shared__ float sNrm[16];

  stage_lds_async(We, sW, EDGE_DIM * 128, threadIdx.x, 128);

  const int e0 = blockIdx.x * 16;
  if (threadIdx.x < 16) {
    int e = e0 + threadIdx.x; if (e >= E) e = E - 1;
    int r = row[e], cc = col[e];
    sRow[threadIdx.x] = r;
    sCol[threadIdx.x] = cc;
    sNrm[threadIdx.x] = dinv[r] * dinv[cc];
  }
  __syncthreads();

  const int wave = threadIdx.x >> 5;
  const int lane = threadIdx.x & 31;
  const int mrow = lane & 15;
  const int hi   = lane >> 4;
  const int kh   = hi * 2;
  const int n0   = wave * 32;
  const int ncol = lane & 15;

  int ei = e0 + mrow; if (ei >= E) ei = E - 1;
  const float* __restrict__ arow = edge_attr + (size_t)ei * EDGE_DIM;

  v8f c0 = {}; v8f c1 = {};
#pragma unroll 4
  for (int k = 0; k < EDGE_DIM; k += 4) {
    v2f a = *(const v2f*)(arow + k + kh);
    v2f b0, b1;
    b0.x = sW[(k + kh    ) * 128 + n0 + ncol];
    b0.y = sW[(k + kh + 1) * 128 + n0 + ncol];
    b1.x = sW[(k + kh    ) * 128 + n0 + 16 + ncol];
    b1.y = sW[(k + kh + 1) * 128 + n0 + 16 + ncol];
    c0 = wmma4(a, b0, c0);
    c1 = wmma4(a, b1, c1);
  }

  const int ch0 = n0 + ncol, ch1 = n0 + 16 + ncol;
#pragma unroll
  for (int v = 0; v < 8; ++v) {
    int m = v + (hi << 3);               // edge index within tile
    if (e0 + m >= E) continue;
    int r = sRow[m], cc = sCol[m];
    float nrm = sNrm[m];
    float v0 = (c0[v] + hpre[(size_t)r * 128 + ch0]) * nrm;
    float v1 = (c1[v] + hpre[(size_t)r * 128 + ch1]) * nrm;
    atomicAdd(&acc[(size_t)cc * 128 + ch0], v0);
    atomicAdd(&acc[(size_t)cc * 128 + ch1], v1);
  }
}

// ---------------- residual + LayerNorm (one wave32 per node row) ----------------
__global__ __launch_bounds__(128) void k_out_ln(const float* __restrict__ outp,
                                                const float* __restrict__ x,
                                                const float* __restrict__ b_out,
                                                const float* __restrict__ g,
                                                const float* __restrict__ b,
                                                float* __restrict__ y, int n) {
  int rowid = blockIdx.x * (blockDim.x >> 5) + (threadIdx.x >> 5);
  if (rowid >= n) return;
  int lane = threadIdx.x & 31;
  float t[4];
  float s = 0.f;
#pragma unroll
  for (int i = 0; i < 4; ++i) {
    int c = lane + 32 * i;
    t[i] = outp[(size_t)rowid * 128 + c] + b_out[c] + x[(size_t)rowid * 128 + c];
    s += t[i];
  }
  for (int off = 16; off > 0; off >>= 1) s += __shfl_xor(s, off, 32);
  float mu = s * (1.0f / 128.0f);
  float var = 0.f;
#pragma unroll
  for (int i = 0; i < 4; ++i) { float d = t[i] - mu; var += d * d; }
  for (int off = 16; off > 0; off >>= 1) var += __shfl_xor(var, off, 32);
  var *= (1.0f / 128.0f);
  float rs = rsqrtf(var + 1e-5f);
#pragma unroll
  for (int i = 0; i < 4; ++i) {
    int c = lane + 32 * i;
    y[(size_t)rowid * 128 + c] = (t[i] - mu) * rs * g[c] + b[c];
  }
}

// ---------------- launcher ----------------
extern "C" void kernel_launch(void* const* d_in, const int* in_sizes, int n_in,
                              void* d_out, int out_size, void* d_ws, size_t ws_size,
                              hipStream_t stream) {
  const float* x         = (const float*)d_in[0];
  const int*   eidx      = (const int*)d_in[1];
  const float* edge_attr = (const float*)d_in[2];
  // d_in[3..6]: W_qkv, b_qkv, W_ek, b_ek -> dead code in the reference, skipped.
  const float* W_n1  = (const float*)d_in[7];
  const float* W_e1  = (const float*)d_in[8];
  const float* b1    = (const float*)d_in[9];
  const float* W_n2  = (const float*)d_in[10];
  const float* W_e2  = (const float*)d_in[11];
  const float* b2    = (const float*)d_in[12];
  const float* W_out = (const float*)d_in[13];
  const float* b_out = (const float*)d_in[14];
  const float* ln_g  = (const float*)d_in[15];
  const float* ln_b  = (const float*)d_in[16];

  const int n = in_sizes[0] / NODE_DIM;       // 100000
  const int E = in_sizes[1] / 2;              // 800000
  const int* row = eidx;
  const int* col = eidx + E;

  float* ws   = (float*)d_ws;
  float* dinv = ws;                           // n floats
  float* buf1 = dinv + n;                     // n*128 floats
  float* buf2 = buf1 + (size_t)n * 128;       // n*128 floats

  const int T = 256;
  const int nodeTiles = (n + 15) / 16;        // 6250 exact
  const int edgeTiles = (E + 15) / 16;        // 50000 exact
  const int totalNC   = n * 128;
  const int totalNC4  = totalNC / 4;

  // degree -> dinv (in place)
  k_set1<<<(n + T - 1) / T, T, 0, stream>>>(dinv, n);
  k_count_deg<<<(E + T - 1) / T, T, 0, stream>>>(col, dinv, E);
  k_rsqrt_inplace<<<(n + T - 1) / T, T, 0, stream>>>(dinv, n);

  // ---- layer 1 ----
  k_gemm_wmma<128><<<nodeTiles, 128, 0, stream>>>(x, W_n1, buf1, n);        // hpre1
  k_init_acc4<<<(totalNC4 + T - 1) / T, T, 0, stream>>>(buf1, dinv, b1, buf2, totalNC4);
  k_edge_msg<<<edgeTiles, 128, 0, stream>>>(edge_attr, W_e1, row, col, buf1, dinv, buf2, E);
  k_gelu_inplace4<<<(totalNC4 + T - 1) / T, T, 0, stream>>>(buf2, totalNC4); // buf2 = h1

  // ---- layer 2 ----
  k_gemm_wmma<128><<<nodeTiles, 128, 0, stream>>>(buf2, W_n2, buf1, n);     // hpre2
  k_init_acc4<<<(totalNC4 + T - 1) / T, T, 0, stream>>>(buf1, dinv, b2, buf2, totalNC4);
  k_edge_msg<<<edgeTiles, 128, 0, stream>>>(edge_attr, W_e2, row, col, buf1, dinv, buf2, E);
  k_gelu_inplace4<<<(totalNC4 + T - 1) / T, T, 0, stream>>>(buf2, totalNC4); // buf2 = h2

  // ---- output projection + residual + LayerNorm ----
  k_gemm_wmma<128><<<nodeTiles, 128, 0, stream>>>(buf2, W_out, buf1, n);    // outp
  k_out_ln<<<(n + 3) / 4, 128, 0, stream>>>(buf1, x, b_out, ln_g, ln_b, (float*)d_out, n);
}